// TreeLSTMCellDP_80229989089608
// MI455X (gfx1250) — compile-verified
//
#include <hip/hip_runtime.h>

typedef float v2f __attribute__((ext_vector_type(2)));
typedef float v8f __attribute__((ext_vector_type(8)));

#define Hh 128
#define NROWS 16
#define JT 16
#define NJT (Hh / JT)          // 8
#define KHALF 128
#define NSL 6
#define NT 3
#define A_SL 2048              // KHALF*NROWS floats per (type,slice)
#define A_TOTAL (NT * NSL * A_SL)        // 36864 floats
#define Z_TILE 256
#define Z_TOTAL (NT * NSL * 5 * Z_TILE)  // 23040 floats
#define LDS_FLOATS (A_TOTAL + Z_TOTAL)   // 59904 floats = 239616 B

__device__ __forceinline__ float sigmf_(float x) { return 1.0f / (1.0f + __expf(-x)); }
__device__ __forceinline__ float min4_(float a, float b, float c, float d) {
    return fminf(fminf(a, b), fminf(c, d));
}
__device__ __forceinline__ float max4_(float a, float b, float c, float d) {
    return fmaxf(fmaxf(a, b), fmaxf(c, d));
}
// interval multiply: (av,al,au) * (bv,bl,bu) -> (ov,ol,ou)
__device__ __forceinline__ void imul3_(float av, float al, float au,
                                       float bv, float bl, float bu,
                                       float& ov, float& ol, float& ou) {
    ov = av * bv;
    float ll = al * bl, lu = al * bu, ul = au * bl, uu = au * bu;
    ol = min4_(ll, lu, ul, uu);
    ou = max4_(ll, lu, ul, uu);
}

__global__ __launch_bounds__(320, 1) void treelstm_ibp_fused(
    const float* __restrict__ h_val, const float* __restrict__ h_lb, const float* __restrict__ h_ub,
    const float* __restrict__ c_val, const float* __restrict__ c_lb, const float* __restrict__ c_ub,
    const float* __restrict__ W_uf, const float* __restrict__ b_uf,
    const float* __restrict__ W_uiou, const float* __restrict__ b_iou,
    const int* __restrict__ unk, float* __restrict__ out, int Nn) {
    extern __shared__ float lds[];
    float* Ab = lds;              // swizzled WMMA-A fragments: [type][slice]
    float* Zb = lds + A_TOTAL;    // z results per j-tile: [type][slice][group][row][col]

    const int tid = threadIdx.x;
    const int n0 = blockIdx.x * NROWS;

    // ---------------- Phase 1: load h, build v / mu / r A-fragments in LDS ----
    // global h layout: n*768 + side*384 + l0*128 + k  == n*768 + s*128 + k (s = side*3+l0)
    for (int e = tid; e < NROWS * NSL * KHALF; e += (int)blockDim.x) {
        int m = e / (NSL * KHALF);
        int rem = e - m * (NSL * KHALF);
        int s = rem >> 7;
        int k = rem & 127;
        size_t g = (size_t)(n0 + m) * 768 + (size_t)rem;
        float v = h_val[g];
        float l = h_lb[g];
        float u = h_ub[g];
        float mu = 0.5f * (l + u);
        float r = 0.5f * (u - l);
        // WMMA f32 16x16x4 A-fragment swizzle: element (k,m) -> lane hi*16+m, vgpr k&1
        int off = ((k >> 2) << 6) + (((k >> 1) & 1) << 5) + (m << 1) + (k & 1);
        int sb = s * A_SL + off;
        Ab[0 * NSL * A_SL + sb] = v;
        Ab[1 * NSL * A_SL + sb] = mu;
        Ab[2 * NSL * A_SL + sb] = r;
    }
    __syncthreads();

    const int wave = tid >> 5;
    const int lane = tid & 31;
    const int lm = lane & 15;
    const int hi = lane >> 4;
    const float PINF = __builtin_inff();

    for (int jt = 0; jt < NJT; ++jt) {
        const int j0 = jt * JT;

        // ---------------- Phase 2: WMMA f32 GEMM ----------------------------
        // 10 waves: wave = group(0..4)*2 + side. group cols = g*128 + j0 + n.
        if (wave < 10) {
            const int g = wave >> 1;
            const int side = wave & 1;
            const float* Wsel = (g < 2) ? W_uf : W_uiou;
            const int row0 = ((g < 2) ? g * Hh : (g - 2) * Hh) + j0;
            // per-lane B base: row = row0 + n(=lm), K offset side*128 + 2*hi (+ kk*4 + v)
            const float* bptr = Wsel + (size_t)(row0 + lm) * 256 + side * Hh + hi * 2;

            v8f acc[NT][3] = {};
            for (int kk = 0; kk < 32; ++kk) {
                v2f b = *(const v2f*)(bptr + kk * 4);
                v2f ba;
                ba.x = fabsf(b.x);
                ba.y = fabsf(b.y);
#pragma unroll
                for (int t = 0; t < NT; ++t) {
                    v2f bb = (t == 2) ? ba : b;
#pragma unroll
                    for (int ls = 0; ls < 3; ++ls) {
                        const int s = side * 3 + ls;
                        v2f a = *(const v2f*)(Ab + (t * NSL + s) * A_SL + kk * 64 + lane * 2);
                        acc[t][ls] = __builtin_amdgcn_wmma_f32_16x16x4_f32(
                            false, a, false, bb, (short)0, acc[t][ls], false, false);
                    }
                }
            }
            // store accumulators to Zb as [row(M=sample)][col(N=j within tile)]
#pragma unroll
            for (int t = 0; t < NT; ++t)
#pragma unroll
                for (int ls = 0; ls < 3; ++ls) {
                    const int s = side * 3 + ls;
                    float* zp = Zb + ((t * NSL + s) * 5 + g) * Z_TILE;
                    v8f a = acc[t][ls];
#pragma unroll
                    for (int v = 0; v < 8; ++v) zp[(v + 8 * hi) * 16 + lm] = a[v];
                }
        }
        __syncthreads();

        // ---------------- Phase 3: interval epilogue -------------------------
        if (tid < 256) {
            const int jl = tid & 15;
            const int rw = tid >> 4;
            const int n = n0 + rw;
            const int j = j0 + jl;

            float z[NT][NSL][5];
#pragma unroll
            for (int t = 0; t < NT; ++t)
#pragma unroll
                for (int s = 0; s < NSL; ++s)
#pragma unroll
                    for (int g = 0; g < 5; ++g)
                        z[t][s][g] = Zb[((t * NSL + s) * 5 + g) * Z_TILE + rw * 16 + jl];

            const size_t gb = (size_t)n * 768 + (size_t)j;
            float cv[6], clb_[6], cub_[6], hlb_[6], hub_[6];
#pragma unroll
            for (int s = 0; s < 6; ++s) {
                size_t ix = gb + (size_t)s * 128;
                cv[s] = c_val[ix];
                clb_[s] = c_lb[ix];
                cub_[s] = c_ub[ix];
                hlb_[s] = h_lb[ix];
                hub_[s] = h_ub[ix];
            }
            const float bf0 = b_uf[j], bf1 = b_uf[Hh + j];
            const float bio[3] = {b_iou[j], b_iou[Hh + j], b_iou[2 * Hh + j]};
            const int m0 = unk[2 * n], m1 = unk[2 * n + 1];

            float iouV[3][3], iouL[3][3], iouU[3][3];
            float pcV[3], pcL[3], pcU[3];
            const int pd0[6] = {0, 1, 1, 2, 2, 2};
            const int pl0[6] = {0, 0, 1, 0, 1, 2};
#pragma unroll
            for (int p = 0; p < 6; ++p) {
                const int d0 = pd0[p], l0 = pl0[p];
                const int sL = l0, sR = 3 + (d0 - l0);
                float zv[5], zlo[5], zup[5];
#pragma unroll
                for (int g = 0; g < 5; ++g) {
                    float bb = (g == 0) ? bf0 : ((g == 1) ? bf1 : 0.0f);
                    float v = z[0][sL][g] + z[0][sR][g] + bb;
                    float mm = z[1][sL][g] + z[1][sR][g] + bb;
                    float rr = z[2][sL][g] + z[2][sR][g];
                    zv[g] = v;
                    zlo[g] = mm - rr;
                    zup[g] = mm + rr;
                }
                float f0v = sigmf_(zv[0]), f0l = sigmf_(zlo[0]), f0u = sigmf_(zup[0]);
                float f1v = sigmf_(zv[1]), f1l = sigmf_(zlo[1]), f1u = sigmf_(zup[1]);
                float aV, aL, aU, bV, bL, bU;
                imul3_(f0v, f0l, f0u, cv[sL], clb_[sL], cub_[sL], aV, aL, aU);
                imul3_(f1v, f1l, f1u, cv[sR], clb_[sR], cub_[sR], bV, bL, bU);
                float ccV = aV + bV, ccL = aL + bL, ccU = aU + bU;
                if (l0 == 0) {
                    pcV[d0] = ccV;
                    pcL[d0] = ccL;
                    pcU[d0] = ccU;
#pragma unroll
                    for (int q = 0; q < 3; ++q) {
                        iouV[d0][q] = zv[2 + q];
                        iouL[d0][q] = zlo[2 + q];
                        iouU[d0][q] = zup[2 + q];
                    }
                } else {
                    pcL[d0] = fminf(pcL[d0], ccL);
                    pcU[d0] = fmaxf(pcU[d0], ccU);
#pragma unroll
                    for (int q = 0; q < 3; ++q) {
                        iouL[d0][q] = fminf(iouL[d0][q], zlo[2 + q]);
                        iouU[d0][q] = fmaxf(iouU[d0][q], zup[2 + q]);
                    }
                }
            }

            const size_t CS = (size_t)Nn * 384;
#pragma unroll
            for (int d0 = 0; d0 < 3; ++d0) {
                float iV = sigmf_(iouV[d0][0] + bio[0]);
                float iLo = sigmf_(iouL[d0][0] + bio[0]);
                float iUp = sigmf_(iouU[d0][0] + bio[0]);
                float oV = sigmf_(iouV[d0][1] + bio[1]);
                float oLo = sigmf_(iouL[d0][1] + bio[1]);
                float oUp = sigmf_(iouU[d0][1] + bio[1]);
                float uV = tanhf(iouV[d0][2] + bio[2]);
                float uLo = tanhf(iouL[d0][2] + bio[2]);
                float uUp = tanhf(iouU[d0][2] + bio[2]);
                float mV, mLo, mUp;
                imul3_(iV, iLo, iUp, uV, uLo, uUp, mV, mLo, mUp);
                float cnV = mV + pcV[d0], cnL = mLo + pcL[d0], cnU = mUp + pcU[d0];
                float tV = tanhf(cnV), tL = tanhf(cnL), tU = tanhf(cnU);
                float hnV, hnL, hnU;
                imul3_(oV, oLo, oUp, tV, tL, tU, hnV, hnL, hnU);
                // aux merge
                int c0 = m0 > 1 ? m0 : 1;
                c0 = c0 < d0 ? c0 : d0;  // clip(m0,1,d0); d0=0 -> 0
                int c1 = m1 > 1 ? m1 : 1;
                c1 = c1 < d0 ? c1 : d0;
                bool mLk = (m0 > 0) && (m0 <= d0);
                bool mRk = (m1 > 0) && (m1 <= d0);
                int iLx = d0 - c0, iRx = d0 - c1;
                float axhl = fminf(mLk ? hlb_[3 + iLx] : PINF, mRk ? hlb_[iRx] : PINF);
                float axhu = fmaxf(mLk ? hub_[3 + iLx] : -PINF, mRk ? hub_[iRx] : -PINF);
                float axcl = fminf(mLk ? clb_[3 + iLx] : PINF, mRk ? clb_[iRx] : PINF);
                float axcu = fmaxf(mLk ? cub_[3 + iLx] : -PINF, mRk ? cub_[iRx] : -PINF);
                float hoL = fminf(hnL, axhl), hoU = fmaxf(hnU, axhu);
                float coL = fminf(cnL, axcl), coU = fmaxf(cnU, axcu);
                size_t ob = (size_t)n * 384 + (size_t)d0 * 128 + (size_t)j;
                out[ob] = hnV;
                out[CS + ob] = hoL;
                out[2 * CS + ob] = hoU;
                out[3 * CS + ob] = cnV;
                out[4 * CS + ob] = coL;
                out[5 * CS + ob] = coU;
            }
            if (jt == 0 && jl == 0) {
                int nu = (m0 > 0 && m1 > 0) ? (m0 + m1) : 0;
                out[6 * CS + (size_t)n] = (float)nu;
            }
        }
        __syncthreads();
    }
}

extern "C" void kernel_launch(void* const* d_in, const int* in_sizes, int n_in,
                              void* d_out, int out_size, void* d_ws, size_t ws_size,
                              hipStream_t stream) {
    (void)n_in;
    (void)out_size;
    (void)d_ws;
    (void)ws_size;
    const float* h_val = (const float*)d_in[0];
    const float* h_lb = (const float*)d_in[1];
    const float* h_ub = (const float*)d_in[2];
    const float* c_val = (const float*)d_in[3];
    const float* c_lb = (const float*)d_in[4];
    const float* c_ub = (const float*)d_in[5];
    const float* W_uf = (const float*)d_in[6];
    const float* b_uf = (const float*)d_in[7];
    const float* W_uiou = (const float*)d_in[8];
    const float* b_iou = (const float*)d_in[9];
    const int* unk = (const int*)d_in[10];
    float* out = (float*)d_out;

    const int N = in_sizes[0] / 768;  // (N,2,3,1,1,128)
    const size_t shmem = (size_t)LDS_FLOATS * sizeof(float);  // 239,616 B < 320 KB/WGP
    (void)hipFuncSetAttribute((const void*)treelstm_ibp_fused,
                              hipFuncAttributeMaxDynamicSharedMemorySize, (int)shmem);
    treelstm_ibp_fused<<<dim3(N / NROWS), dim3(320), shmem, stream>>>(
        h_val, h_lb, h_ub, c_val, c_lb, c_ub, W_uf, b_uf, W_uiou, b_iou, unk, out, N);
}